// GCNEncoder1_11948599018106
// MI455X (gfx1250) — compile-verified
//
#include <hip/hip_runtime.h>
#include <hip/hip_bf16.h>

typedef __attribute__((ext_vector_type(2))) float v2f;
typedef __attribute__((ext_vector_type(8))) float v8f;

// ---------------------------------------------------------------------------
// Stage 0: agg = x (GIN eps=0 self term), deg = 1 (A+I self loop)
// ---------------------------------------------------------------------------
__global__ void init_node_kernel(const float* __restrict__ x,
                                 float* __restrict__ agg,
                                 float* __restrict__ deg, int N) {
    int idx = blockIdx.x * blockDim.x + threadIdx.x;   // over N*64
    if (idx >= N * 64) return;
    agg[idx] = x[idx];
    if ((idx & 63) == 0) deg[idx >> 6] = 1.0f;
}

// ---------------------------------------------------------------------------
// Stage 1: GIN edge scatter. One wave32 per edge: 64 channels, 2 per lane.
// ---------------------------------------------------------------------------
__global__ void gin_edge_kernel(const long long* __restrict__ ei,
                                const float* __restrict__ x,
                                float* __restrict__ agg,
                                float* __restrict__ deg, int E) {
    int wid  = (blockIdx.x * blockDim.x + threadIdx.x) >> 5;
    int lane = threadIdx.x & 31;
    if (wid >= E) return;
    long long s = ei[wid];       // edge_index[0] = src
    long long d = ei[(long long)E + wid];  // edge_index[1] = dst
    if (lane == 0) atomicAdd(&deg[d], 1.0f);
    const float* xs = x + s * 64;
    float* ad = agg + d * 64;
    atomicAdd(&ad[lane],      xs[lane]);
    atomicAdd(&ad[lane + 32], xs[lane + 32]);
}

__global__ void dinv_kernel(const float* __restrict__ deg,
                            float* __restrict__ dinv, int N) {
    int i = blockIdx.x * blockDim.x + threadIdx.x;
    if (i < N) dinv[i] = rsqrtf(deg[i]);
}

// ---------------------------------------------------------------------------
// Stage 2: h = relu(agg @ W64x64 + b).  WMMA f32 16x16x4, one 16x16 tile/wave.
// Block = 128 threads = 4 waves = 4 column tiles (HID_C = 64).
// ---------------------------------------------------------------------------
__global__ void gin_gemm_kernel(const float* __restrict__ agg,
                                const float* __restrict__ W,   // [64,64]
                                const float* __restrict__ b,   // [64]
                                float* __restrict__ h, int N) {
    int wave = threadIdx.x >> 5;          // col tile 0..3
    int lane = threadIdx.x & 31;
    int row0 = blockIdx.x * 16;
    int col0 = wave * 16;
    int r    = lane & 15;
    int koff = (lane >> 4) * 2;           // lanes 16-31 hold K+2,K+3
    int arow = row0 + r; if (arow >= N) arow = N - 1;   // clamp (N%16==0 normally)

    v8f acc = {};
    #pragma unroll
    for (int kk = 0; kk < 64; kk += 4) {
        v2f a, bf;
        const float* ap = agg + (size_t)arow * 64 + kk + koff;
        a.x = ap[0];
        a.y = ap[1];
        bf.x = W[(size_t)(kk + koff)     * 64 + col0 + r];
        bf.y = W[(size_t)(kk + koff + 1) * 64 + col0 + r];
        acc = __builtin_amdgcn_wmma_f32_16x16x4_f32(
            false, a, false, bf, (short)0, acc, false, false);
    }
    // D layout: VGPR i -> row (lane<16 ? i : i+8), col = col0 + (lane&15)
    int outCol = col0 + r;
    int mbase  = (lane < 16) ? 0 : 8;
    float bias = b[outCol];
    #pragma unroll
    for (int i = 0; i < 8; ++i) {
        int m = row0 + mbase + i;
        if (m < N) {
            float v = acc[i] + bias;
            h[(size_t)m * 64 + outCol] = fmaxf(v, 0.0f);
        }
    }
}

// ---------------------------------------------------------------------------
// Stage 3: xw_mu = h @ mu_W [64,32], xw_lv = h @ lv_W [64,32].
// 4 waves/block: waves 0-1 -> mu col tiles, waves 2-3 -> lv col tiles.
// ---------------------------------------------------------------------------
__global__ void enc_gemm_kernel(const float* __restrict__ h,
                                const float* __restrict__ muW,
                                const float* __restrict__ lvW,
                                float* __restrict__ xw_mu,
                                float* __restrict__ xw_lv, int N) {
    int wave = threadIdx.x >> 5;
    int lane = threadIdx.x & 31;
    const float* W   = (wave < 2) ? muW   : lvW;
    float*       out = (wave < 2) ? xw_mu : xw_lv;
    int row0 = blockIdx.x * 16;
    int col0 = (wave & 1) * 16;
    int r    = lane & 15;
    int koff = (lane >> 4) * 2;
    int arow = row0 + r; if (arow >= N) arow = N - 1;

    v8f acc = {};
    #pragma unroll
    for (int kk = 0; kk < 64; kk += 4) {
        v2f a, bf;
        const float* ap = h + (size_t)arow * 64 + kk + koff;
        a.x = ap[0];
        a.y = ap[1];
        bf.x = W[(size_t)(kk + koff)     * 32 + col0 + r];
        bf.y = W[(size_t)(kk + koff + 1) * 32 + col0 + r];
        acc = __builtin_amdgcn_wmma_f32_16x16x4_f32(
            false, a, false, bf, (short)0, acc, false, false);
    }
    int outCol = col0 + r;
    int mbase  = (lane < 16) ? 0 : 8;
    #pragma unroll
    for (int i = 0; i < 8; ++i) {
        int m = row0 + mbase + i;
        if (m < N) out[(size_t)m * 32 + outCol] = acc[i];
    }
}

// ---------------------------------------------------------------------------
// Stage 4: GCN edge scatter for mu and lv simultaneously, wave per edge.
// Accumulates directly into d_out (zeroed beforehand).
// ---------------------------------------------------------------------------
__global__ void gcn_edge_kernel(const long long* __restrict__ ei,
                                const float* __restrict__ xw_mu,
                                const float* __restrict__ xw_lv,
                                const float* __restrict__ dinv,
                                float* __restrict__ acc_mu,
                                float* __restrict__ acc_lv, int E) {
    int wid  = (blockIdx.x * blockDim.x + threadIdx.x) >> 5;
    int lane = threadIdx.x & 31;
    if (wid >= E) return;
    long long s = ei[wid];
    long long d = ei[(long long)E + wid];
    float norm = dinv[s] * dinv[d];
    atomicAdd(&acc_mu[d * 32 + lane], xw_mu[s * 32 + lane] * norm);
    atomicAdd(&acc_lv[d * 32 + lane], xw_lv[s * 32 + lane] * norm);
}

// ---------------------------------------------------------------------------
// Stage 5: self-loop term + bias; ReLU on mu. In place on d_out.
// ---------------------------------------------------------------------------
__global__ void finalize_kernel(const float* __restrict__ xw_mu,
                                const float* __restrict__ xw_lv,
                                const float* __restrict__ dinv,
                                const float* __restrict__ mu_b,
                                const float* __restrict__ lv_b,
                                float* __restrict__ out_mu,
                                float* __restrict__ out_lv, int N) {
    int idx = blockIdx.x * blockDim.x + threadIdx.x;   // over N*32
    if (idx >= N * 32) return;
    int i = idx >> 5, c = idx & 31;
    float di = dinv[i];
    float d2 = di * di;
    float m = out_mu[idx] + xw_mu[idx] * d2 + mu_b[c];
    out_mu[idx] = fmaxf(m, 0.0f);
    out_lv[idx] = out_lv[idx] + xw_lv[idx] * d2 + lv_b[c];
}

// ---------------------------------------------------------------------------
extern "C" void kernel_launch(void* const* d_in, const int* in_sizes, int n_in,
                              void* d_out, int out_size, void* d_ws, size_t ws_size,
                              hipStream_t stream) {
    const float*     x    = (const float*)d_in[0];
    const long long* ei   = (const long long*)d_in[1];   // int64 [2, E]
    const float*     ginW = (const float*)d_in[2];
    const float*     ginb = (const float*)d_in[3];
    const float*     muW  = (const float*)d_in[4];
    const float*     mub  = (const float*)d_in[5];
    const float*     lvW  = (const float*)d_in[6];
    const float*     lvb  = (const float*)d_in[7];

    const int N = in_sizes[0] / 64;     // 100000
    const int E = in_sizes[1] / 2;      // 1600000

    float* ws    = (float*)d_ws;
    float* agg   = ws;                                   // N*64
    float* h     = agg   + (size_t)N * 64;               // N*64
    float* xw_mu = h     + (size_t)N * 64;               // N*32
    float* xw_lv = xw_mu + (size_t)N * 32;               // N*32
    float* deg   = xw_lv + (size_t)N * 32;               // N
    float* dinv  = deg   + (size_t)N;                    // N

    float* out_mu = (float*)d_out;                       // N*32
    float* out_lv = out_mu + (size_t)N * 32;             // N*32

    // Stage 0: self terms
    init_node_kernel<<<(N * 64 + 255) / 256, 256, 0, stream>>>(x, agg, deg, N);

    // Stage 1: GIN edge aggregation (wave per edge, 8 waves/block)
    gin_edge_kernel<<<(E + 7) / 8, 256, 0, stream>>>(ei, x, agg, deg, E);
    dinv_kernel<<<(N + 255) / 256, 256, 0, stream>>>(deg, dinv, N);

    // Stage 2: GIN GEMM + ReLU  (WMMA f32 16x16x4)
    gin_gemm_kernel<<<(N + 15) / 16, 128, 0, stream>>>(agg, ginW, ginb, h, N);

    // Stage 3: mu / lv feature transforms (WMMA)
    enc_gemm_kernel<<<(N + 15) / 16, 128, 0, stream>>>(h, muW, lvW, xw_mu, xw_lv, N);

    // Stage 4: GCN normalized edge scatter into d_out (zero it first)
    hipMemsetAsync(d_out, 0, (size_t)out_size * sizeof(float), stream);
    gcn_edge_kernel<<<(E + 7) / 8, 256, 0, stream>>>(ei, xw_mu, xw_lv, dinv,
                                                     out_mu, out_lv, E);

    // Stage 5: self-loop + bias + ReLU(mu)
    finalize_kernel<<<(N * 32 + 255) / 256, 256, 0, stream>>>(
        xw_mu, xw_lv, dinv, mub, lvb, out_mu, out_lv, N);
}